// GraphTransformer_14276471292549
// MI455X (gfx1250) — compile-verified
//
#include <hip/hip_runtime.h>
#include <cstdint>

typedef __attribute__((ext_vector_type(2))) float v2f;
typedef __attribute__((ext_vector_type(8))) float v8f;

#define WAVE 32

// ---- order-preserving float<->int mapping for atomic max on signed ints ----
__device__ __forceinline__ int f2ord(float f) {
  int b = __float_as_int(f);
  return b >= 0 ? b : (b ^ 0x7FFFFFFF);
}
__device__ __forceinline__ float ord2f(int o) {
  int b = o >= 0 ? o : (o ^ 0x7FFFFFFF);
  return __int_as_float(b);
}

// ---------------------------------------------------------------------------
// fills
// ---------------------------------------------------------------------------
__global__ void fill_f32(float* __restrict__ p, float v, long n) {
  long i = (long)blockIdx.x * blockDim.x + threadIdx.x;
  if (i < n) p[i] = v;
}
__global__ void fill_i32(int* __restrict__ p, int v, long n) {
  long i = (long)blockIdx.x * blockDim.x + threadIdx.x;
  if (i < n) p[i] = v;
}

// ---------------------------------------------------------------------------
// C[M,N] = A[M,K] @ W[K,N] + bias  (all fp32, row-major)
// One wave computes one 16x16 tile of C via V_WMMA_F32_16X16X4_F32.
// A frag layout : lanes 0-15 hold rows M=l, K = k+0 (v0), k+1 (v1);
//                 lanes 16-31 hold K = k+2, k+3.
// B frag layout : lanes 0-15 hold cols N=l, K = k+0 (v0), k+1 (v1);
//                 lanes 16-31 hold K = k+2, k+3.
// C layout      : vgpr r -> row tm + half*8 + r, col tn + l.
// ---------------------------------------------------------------------------
__global__ void gemm_f32_wmma(const float* __restrict__ A,
                              const float* __restrict__ W,
                              const float* __restrict__ bias,
                              float* __restrict__ C,
                              int M, int K, int N) {
  const int wavesPerBlock = blockDim.x / WAVE;
  const int wid  = blockIdx.x * wavesPerBlock + (threadIdx.x / WAVE);
  const int lane = threadIdx.x & (WAVE - 1);
  const int tilesN = N >> 4;
  const int tm = (wid / tilesN) << 4;
  const int tn = (wid % tilesN) << 4;
  if (tm >= M) return;
  const int half = lane >> 4;      // 0: K lo pair, 1: K hi pair
  const int l    = lane & 15;
  int row = tm + l; if (row >= M) row = M - 1;   // safe read clamp
  const float* arow = A + (size_t)row * K + 2 * half;
  const float* wcol = W + (size_t)(2 * half) * N + tn + l;

  v8f acc = {};
  for (int k = 0; k < K; k += 4) {
    v2f a = *(const v2f*)(arow + k);
    v2f b;
    b.x = wcol[(size_t)k * N];
    b.y = wcol[(size_t)(k + 1) * N];
    acc = __builtin_amdgcn_wmma_f32_16x16x4_f32(false, a, false, b,
                                                (short)0, acc, false, false);
  }
  const float bv = bias[tn + l];
  #pragma unroll
  for (int r = 0; r < 8; ++r) {
    int crow = tm + half * 8 + r;
    if (crow < M) C[(size_t)crow * N + tn + l] = acc[r] + bv;
  }
}

// ---------------------------------------------------------------------------
// Layer-1 edge kernels: heads=4, dch=64, feature stride 256.
// One wave per edge; lane owns 8 contiguous floats; head = lane/8.
// ---------------------------------------------------------------------------
__global__ void edge_alpha_h4(const float* __restrict__ q,
                              const float* __restrict__ kk,
                              const int* __restrict__ ei, int E,
                              float* __restrict__ alpha, int* __restrict__ mmax) {
  const int wid = blockIdx.x * (blockDim.x / WAVE) + threadIdx.x / WAVE;
  if (wid >= E) return;
  const int lane = threadIdx.x & (WAVE - 1);
  const int src = ei[wid];
  const int dst = ei[E + wid];
  const float4* qp = (const float4*)(q  + (size_t)dst * 256) + lane * 2;
  const float4* kp = (const float4*)(kk + (size_t)src * 256) + lane * 2;
  float4 q0 = qp[0], q1 = qp[1];
  float4 k0 = kp[0], k1 = kp[1];
  float p = q0.x*k0.x + q0.y*k0.y + q0.z*k0.z + q0.w*k0.w
          + q1.x*k1.x + q1.y*k1.y + q1.z*k1.z + q1.w*k1.w;
  p += __shfl_xor(p, 1, WAVE);
  p += __shfl_xor(p, 2, WAVE);
  p += __shfl_xor(p, 4, WAVE);
  if ((lane & 7) == 0) {
    const int h = lane >> 3;
    const float a = p * 0.125f;  // 1/sqrt(64)
    alpha[(size_t)wid * 4 + h] = a;
    atomicMax(mmax + (size_t)dst * 4 + h, f2ord(a));
  }
}

__global__ void edge_msg_h4(const float* __restrict__ v,
                            const float* __restrict__ alpha,
                            const int* __restrict__ mmax,
                            const int* __restrict__ ei, int E,
                            float* __restrict__ ssum, float* __restrict__ acc) {
  const int wid = blockIdx.x * (blockDim.x / WAVE) + threadIdx.x / WAVE;
  if (wid >= E) return;
  const int lane = threadIdx.x & (WAVE - 1);
  const int src = ei[wid];
  const int dst = ei[E + wid];
  const int h = lane >> 3;
  const float a = alpha[(size_t)wid * 4 + h];
  const float m = ord2f(mmax[(size_t)dst * 4 + h]);
  const float e = __expf(a - m);
  if ((lane & 7) == 0) atomicAdd(ssum + (size_t)dst * 4 + h, e);
  const float4* vp = (const float4*)(v + (size_t)src * 256) + lane * 2;
  float4 v0 = vp[0], v1 = vp[1];
  float* op = acc + (size_t)dst * 256 + lane * 8;
  atomicAdd(op + 0, e * v0.x); atomicAdd(op + 1, e * v0.y);
  atomicAdd(op + 2, e * v0.z); atomicAdd(op + 3, e * v0.w);
  atomicAdd(op + 4, e * v1.x); atomicAdd(op + 5, e * v1.y);
  atomicAdd(op + 6, e * v1.z); atomicAdd(op + 7, e * v1.w);
}

// ---------------------------------------------------------------------------
// Layer-2 edge kernels: heads=1, dch=64, feature stride 64. Lane owns 2 floats.
// ---------------------------------------------------------------------------
__global__ void edge_alpha_h1(const float* __restrict__ q,
                              const float* __restrict__ kk,
                              const int* __restrict__ ei, int E,
                              float* __restrict__ alpha, int* __restrict__ mmax) {
  const int wid = blockIdx.x * (blockDim.x / WAVE) + threadIdx.x / WAVE;
  if (wid >= E) return;
  const int lane = threadIdx.x & (WAVE - 1);
  const int src = ei[wid];
  const int dst = ei[E + wid];
  const float2* qp = (const float2*)(q  + (size_t)dst * 64) + lane;
  const float2* kp = (const float2*)(kk + (size_t)src * 64) + lane;
  float2 qv = qp[0], kv = kp[0];
  float p = qv.x * kv.x + qv.y * kv.y;
  p += __shfl_xor(p, 1, WAVE);
  p += __shfl_xor(p, 2, WAVE);
  p += __shfl_xor(p, 4, WAVE);
  p += __shfl_xor(p, 8, WAVE);
  p += __shfl_xor(p, 16, WAVE);
  if (lane == 0) {
    const float a = p * 0.125f;
    alpha[wid] = a;
    atomicMax(mmax + dst, f2ord(a));
  }
}

__global__ void edge_msg_h1(const float* __restrict__ v,
                            const float* __restrict__ alpha,
                            const int* __restrict__ mmax,
                            const int* __restrict__ ei, int E,
                            float* __restrict__ ssum, float* __restrict__ acc) {
  const int wid = blockIdx.x * (blockDim.x / WAVE) + threadIdx.x / WAVE;
  if (wid >= E) return;
  const int lane = threadIdx.x & (WAVE - 1);
  const int src = ei[wid];
  const int dst = ei[E + wid];
  const float a = alpha[wid];
  const float m = ord2f(mmax[dst]);
  const float e = __expf(a - m);
  if (lane == 0) atomicAdd(ssum + dst, e);
  const float2* vp = (const float2*)(v + (size_t)src * 64) + lane;
  float2 vv = vp[0];
  float* op = acc + (size_t)dst * 64 + lane * 2;
  atomicAdd(op + 0, e * vv.x);
  atomicAdd(op + 1, e * vv.y);
}

// ---------------------------------------------------------------------------
// inout[i] = relu(inout[i] + acc[i] / s[node,head]); s==0 -> no message term.
// HD = heads*dch per node, D = dch.
// ---------------------------------------------------------------------------
__global__ void node_finalize(const float* __restrict__ acc,
                              const float* __restrict__ ssum,
                              float* __restrict__ inout, long total, int HD, int D) {
  long i = (long)blockIdx.x * blockDim.x + threadIdx.x;
  if (i >= total) return;
  int n = (int)(i / HD);
  int h = (int)((i % HD) / D);
  float s = ssum[(size_t)n * (HD / D) + h];
  float inv = s > 0.f ? 1.f / s : 0.f;
  float val = inout[i] + acc[i] * inv;
  inout[i] = val > 0.f ? val : 0.f;
}

// ---------------------------------------------------------------------------
// out[n, 0:2] = h[n, 0:64] @ Wl[64,2] + bl  — one wave per node
// ---------------------------------------------------------------------------
__global__ void final_linear(const float* __restrict__ h,
                             const float* __restrict__ Wl,
                             const float* __restrict__ bl,
                             float* __restrict__ out, int N) {
  const int wid = blockIdx.x * (blockDim.x / WAVE) + threadIdx.x / WAVE;
  if (wid >= N) return;
  const int lane = threadIdx.x & (WAVE - 1);
  float x0 = h[(size_t)wid * 64 + lane];
  float x1 = h[(size_t)wid * 64 + 32 + lane];
  float p0 = x0 * Wl[lane * 2 + 0] + x1 * Wl[(lane + 32) * 2 + 0];
  float p1 = x0 * Wl[lane * 2 + 1] + x1 * Wl[(lane + 32) * 2 + 1];
  p0 += __shfl_xor(p0, 1, WAVE);  p1 += __shfl_xor(p1, 1, WAVE);
  p0 += __shfl_xor(p0, 2, WAVE);  p1 += __shfl_xor(p1, 2, WAVE);
  p0 += __shfl_xor(p0, 4, WAVE);  p1 += __shfl_xor(p1, 4, WAVE);
  p0 += __shfl_xor(p0, 8, WAVE);  p1 += __shfl_xor(p1, 8, WAVE);
  p0 += __shfl_xor(p0, 16, WAVE); p1 += __shfl_xor(p1, 16, WAVE);
  if (lane == 0) {
    out[(size_t)wid * 2 + 0] = p0 + bl[0];
    out[(size_t)wid * 2 + 1] = p1 + bl[1];
  }
}

// ---------------------------------------------------------------------------
extern "C" void kernel_launch(void* const* d_in, const int* in_sizes, int n_in,
                              void* d_out, int out_size, void* d_ws, size_t ws_size,
                              hipStream_t stream) {
  const float* x   = (const float*)d_in[0];
  const int*   ei  = (const int*)d_in[1];
  const float* Wq1 = (const float*)d_in[2];  const float* bq1 = (const float*)d_in[3];
  const float* Wk1 = (const float*)d_in[4];  const float* bk1 = (const float*)d_in[5];
  const float* Wv1 = (const float*)d_in[6];  const float* bv1 = (const float*)d_in[7];
  const float* Ws1 = (const float*)d_in[8];  const float* bs1 = (const float*)d_in[9];
  const float* Wq2 = (const float*)d_in[10]; const float* bq2 = (const float*)d_in[11];
  const float* Wk2 = (const float*)d_in[12]; const float* bk2 = (const float*)d_in[13];
  const float* Wv2 = (const float*)d_in[14]; const float* bv2 = (const float*)d_in[15];
  const float* Ws2 = (const float*)d_in[16]; const float* bs2 = (const float*)d_in[17];
  const float* Wl  = (const float*)d_in[18]; const float* bl  = (const float*)d_in[19];

  const int N = in_sizes[0] / 64;   // 50000
  const int E = in_sizes[1] / 2;    // 800000

  // ---- workspace layout (floats); layer-2 buffers overlay layer-1 buffers
  float* ws = (float*)d_ws;
  size_t o = 0;
  float* q1     = ws + o; o += (size_t)N * 256;
  float* k1     = ws + o; o += (size_t)N * 256;
  float* v1     = ws + o; o += (size_t)N * 256;
  float* h1     = ws + o; o += (size_t)N * 256;   // skip -> layer1 output
  float* acc1   = ws + o; o += (size_t)N * 256;
  float* alpha1 = ws + o; o += (size_t)E * 4;
  int*   m1     = (int*)(ws + o); o += (size_t)N * 4;
  float* s1     = ws + o; o += (size_t)N * 4;
  // layer-2 overlays (valid once layer-1 edge passes are done; stream-ordered)
  float* q2     = q1;
  float* k2     = q1 + (size_t)N * 64;
  float* v2     = q1 + (size_t)N * 128;
  float* h2     = q1 + (size_t)N * 192;           // skip -> layer2 output
  float* acc2   = k1;
  float* alpha2 = v1;
  int*   m2     = m1;
  float* s2     = s1;

  const int TB = 256;                       // 8 waves / block
  const int edgeBlocks = (E + 7) / 8;
  const int NEG_INF_ORD = (int)0x80000000;  // f2ord(-inf) lower bound

  // ---- layer 1 init
  fill_f32<<<(dim3)(((long)N * 256 + TB - 1) / TB), TB, 0, stream>>>(acc1, 0.f, (long)N * 256);
  fill_f32<<<(dim3)(((long)N * 4 + TB - 1) / TB), TB, 0, stream>>>(s1, 0.f, (long)N * 4);
  fill_i32<<<(dim3)(((long)N * 4 + TB - 1) / TB), TB, 0, stream>>>(m1, NEG_INF_ORD, (long)N * 4);

  // ---- layer 1 projections (q,k,v,skip) : [N,64] x [64,256]
  {
    int tiles = ((N + 15) / 16) * (256 / 16);
    int blocks = (tiles + 7) / 8;
    gemm_f32_wmma<<<blocks, TB, 0, stream>>>(x, Wq1, bq1, q1, N, 64, 256);
    gemm_f32_wmma<<<blocks, TB, 0, stream>>>(x, Wk1, bk1, k1, N, 64, 256);
    gemm_f32_wmma<<<blocks, TB, 0, stream>>>(x, Wv1, bv1, v1, N, 64, 256);
    gemm_f32_wmma<<<blocks, TB, 0, stream>>>(x, Ws1, bs1, h1, N, 64, 256);
  }

  // ---- layer 1 edge softmax + aggregation
  edge_alpha_h4<<<edgeBlocks, TB, 0, stream>>>(q1, k1, ei, E, alpha1, m1);
  edge_msg_h4<<<edgeBlocks, TB, 0, stream>>>(v1, alpha1, m1, ei, E, s1, acc1);
  node_finalize<<<(dim3)(((long)N * 256 + TB - 1) / TB), TB, 0, stream>>>(
      acc1, s1, h1, (long)N * 256, 256, 64);

  // ---- layer 2 init (after layer-1 consumers of overlaid buffers are queued)
  fill_f32<<<(dim3)(((long)N * 64 + TB - 1) / TB), TB, 0, stream>>>(acc2, 0.f, (long)N * 64);
  fill_f32<<<(dim3)(((long)N + TB - 1) / TB), TB, 0, stream>>>(s2, 0.f, (long)N);
  fill_i32<<<(dim3)(((long)N + TB - 1) / TB), TB, 0, stream>>>(m2, NEG_INF_ORD, (long)N);

  // ---- layer 2 projections : [N,256] x [256,64]
  {
    int tiles = ((N + 15) / 16) * (64 / 16);
    int blocks = (tiles + 7) / 8;
    gemm_f32_wmma<<<blocks, TB, 0, stream>>>(h1, Wq2, bq2, q2, N, 256, 64);
    gemm_f32_wmma<<<blocks, TB, 0, stream>>>(h1, Wk2, bk2, k2, N, 256, 64);
    gemm_f32_wmma<<<blocks, TB, 0, stream>>>(h1, Wv2, bv2, v2, N, 256, 64);
    gemm_f32_wmma<<<blocks, TB, 0, stream>>>(h1, Ws2, bs2, h2, N, 256, 64);
  }

  // ---- layer 2 edge softmax + aggregation
  edge_alpha_h1<<<edgeBlocks, TB, 0, stream>>>(q2, k2, ei, E, alpha2, m2);
  edge_msg_h1<<<edgeBlocks, TB, 0, stream>>>(v2, alpha2, m2, ei, E, s2, acc2);
  node_finalize<<<(dim3)(((long)N * 64 + TB - 1) / TB), TB, 0, stream>>>(
      acc2, s2, h2, (long)N * 64, 64, 64);

  // ---- final linear [N,64] x [64,2]
  final_linear<<<(N + 7) / 8, TB, 0, stream>>>(h2, Wl, bl, (float*)d_out, N);
}